// SkyGaussian_52312701665955
// MI455X (gfx1250) — compile-verified
//
#include <hip/hip_runtime.h>
#include <math.h>

// ---------------------------------------------------------------------------
// SkyGaussian forward for MI455X (gfx1250, wave32, WMMA).
//   pipeline: project 360k points into 8 cameras -> bilinear sample 3ch ->
//   masked mean -> MLP 3->64 (relu) -> 64->6 (tanh) -> concat with exp(scales)
//   and multiply by projection mask.
// MLP runs on V_WMMA_F32_16X16X4_F32 (exact fp32, matches reference dtype).
// ---------------------------------------------------------------------------

typedef float v2f __attribute__((ext_vector_type(2)));
typedef float v8f __attribute__((ext_vector_type(8)));

#define SN   8
#define IMH  544
#define IMW  960
#define HID  64

__device__ __forceinline__ float wshfl(float v, int src) {
    return __shfl(v, src, 32);   // gfx1250 is wave32
}

__global__ __launch_bounds__(256)
void sky_gaussian_kernel(const float* __restrict__ images,   // (1,S,3,H,W)
                         const float* __restrict__ extr,     // (S,4,4)
                         const float* __restrict__ intr,     // (S,3,3)
                         const float* __restrict__ W1,       // (3,64)
                         const float* __restrict__ b1,       // (64)
                         const float* __restrict__ W2,       // (64,6)
                         const float* __restrict__ b2,       // (6)
                         const float* __restrict__ pcd,      // (N,3)
                         const float* __restrict__ scl,      // (N,3)
                         const int*   __restrict__ dsamp,    // scalar
                         float*       __restrict__ out,      // (N,6)
                         int N)
{
    // one 16x64 fp32 h-tile staging buffer per wave (8 waves / 256 threads)
    __shared__ float hbuf[8][16][HID + 4];

    const int  tid  = threadIdx.x;
    const int  wave = tid >> 5;
    const int  lane = tid & 31;
    const bool hi   = lane >= 16;
    const int  lm   = lane & 15;

    const int base = blockIdx.x * 256 + wave * 32;  // 32 points per wave
    const int p    = base + lane;
    const int pc   = (p < N) ? p : (N - 1);

    const float ds     = (float)dsamp[0];
    const float inv_ds = 1.0f / ds;

    // ---- per-lane projection + bilinear sampling over 8 cameras ----------
    const float x  = pcd[(size_t)pc * 3 + 0];
    const float y  = pcd[(size_t)pc * 3 + 1];
    const float z3 = pcd[(size_t)pc * 3 + 2];

    float f0 = 0.f, f1 = 0.f, f2 = 0.f, cnt = 0.f;
    if (p < N) {
#pragma unroll
        for (int s = 0; s < SN; ++s) {
            const float* E = extr + s * 16;
            float cx = E[0]*x + E[1]*y + E[2]*z3  + E[3];
            float cy = E[4]*x + E[5]*y + E[6]*z3  + E[7];
            float cz = E[8]*x + E[9]*y + E[10]*z3 + E[11];
            const float* K = intr + s * 9;
            // Ks = K / D, D = [[ds,1,ds],[1,ds,ds],[1,1,1]]
            float u0 = (K[0]*inv_ds)*cx + K[1]*cy          + (K[2]*inv_ds)*cz;
            float u1 = K[3]*cx          + (K[4]*inv_ds)*cy + (K[5]*inv_ds)*cz;
            float u2 = K[6]*cx          + K[7]*cy          + K[8]*cz;
            float zs = (fabsf(u2) > 1e-6f) ? u2 : 1e-6f;
            float px = u0 / zs, py = u1 / zs;
            bool m = (u2 > 1e-3f) && (px >= 0.f) && (px <= (float)(IMW - 1))
                                  && (py >= 0.f) && (py <= (float)(IMH - 1));
            if (m) {
                float fx0 = floorf(px), fy0 = floorf(py);
                int x0 = (int)fminf(fmaxf(fx0, 0.f), (float)(IMW - 2));
                int y0 = (int)fminf(fmaxf(fy0, 0.f), (float)(IMH - 2));
                float wx = px - (float)x0, wy = py - (float)y0;
                float w00 = (1.f - wx) * (1.f - wy), w01 = wx * (1.f - wy);
                float w10 = (1.f - wx) * wy,         w11 = wx * wy;
                const float* img = images + (size_t)s * 3 * IMH * IMW
                                 + (size_t)y0 * IMW + x0;
                {
                    const float* ic = img;
                    f0 += ic[0]*w00 + ic[1]*w01 + ic[IMW]*w10 + ic[IMW + 1]*w11;
                    ic += (size_t)IMH * IMW;
                    f1 += ic[0]*w00 + ic[1]*w01 + ic[IMW]*w10 + ic[IMW + 1]*w11;
                    ic += (size_t)IMH * IMW;
                    f2 += ic[0]*w00 + ic[1]*w01 + ic[IMW]*w10 + ic[IMW + 1]*w11;
                }
                cnt += 1.f;
            }
        }
    }
    const float anym = (cnt > 0.f) ? 1.f : 0.f;
    const float rc   = 1.f / fmaxf(cnt, 1.f);
    f0 *= rc; f1 *= rc; f2 *= rc;

    // ---- scales half of the output: exp(bg_scales) * proj_mask -----------
    if (p < N) {
#pragma unroll
        for (int j = 0; j < 3; ++j)
            out[(size_t)p * 6 + 3 + j] = expf(scl[(size_t)p * 3 + j]) * anym;
    }

    // ---- preload WMMA B operands -----------------------------------------
    // GEMM1 B tiles: W1 is (3 x 64) row-major, K padded 3 -> 4.
    // B(4x16) layout: lanes 0-15 hold K0/K1 rows, lanes 16-31 hold K2/K3.
    v2f bw1[4];
    float b1v[4];
#pragma unroll
    for (int j = 0; j < 4; ++j) {
        int col  = 16 * j + lm;
        bw1[j].x = hi ? W1[2 * HID + col] : W1[0 * HID + col];
        bw1[j].y = hi ? 0.f               : W1[1 * HID + col];
        b1v[j]   = b1[col];
    }
    // GEMM2 B chunks: W2 is (64 x 6) row-major, N padded 6 -> 16.
    v2f bw2[16];
#pragma unroll
    for (int k = 0; k < 16; ++k) {
        int r0 = 4 * k + (hi ? 2 : 0);
        int r1 = 4 * k + (hi ? 3 : 1);
        bw2[k].x = (lm < 6) ? W2[r0 * 6 + lm] : 0.f;
        bw2[k].y = (lm < 6) ? W2[r1 * 6 + lm] : 0.f;
    }
    const float b2v = (lm < 6) ? b2[lm] : 0.f;

    // ---- MLP: two 16-point tiles per wave --------------------------------
#pragma unroll
    for (int t = 0; t < 2; ++t) {
        // Build A (16x4, K padded): row M sourced from lane 16*t + M.
        int   sl = 16 * t + lm;
        float s0 = wshfl(f0, sl);
        float s1 = wshfl(f1, sl);
        float s2 = wshfl(f2, sl);
        v2f a;
        a.x = hi ? s2  : s0;   // lanes 0-15: K0 ; lanes 16-31: K2
        a.y = hi ? 0.f : s1;   // lanes 0-15: K1 ; lanes 16-31: K3 (pad)

        // GEMM1: h(16x64) = A(16x4) x W1(4x64)   [4 WMMA, one per 16-col tile]
#pragma unroll
        for (int j = 0; j < 4; ++j) {
            v8f c = {};
            c = __builtin_amdgcn_wmma_f32_16x16x4_f32(
                    false, a, false, bw1[j], (short)0, c, false, false);
            int col = 16 * j + lm;
#pragma unroll
            for (int v = 0; v < 8; ++v) {
                int row = v + (hi ? 8 : 0);          // C layout: M = v + 8*hi
                hbuf[wave][row][col] = fmaxf(c[v] + b1v[j], 0.f);  // bias+relu
            }
        }
        // cross-lane RAW through LDS within this wave: drain DS counter
        asm volatile("s_wait_dscnt 0" ::: "memory");

        // GEMM2: out6(16x16) = h(16x64) x W2pad(64x16)   [16 chained WMMA]
        v8f acc = {};
#pragma unroll
        for (int k = 0; k < 16; ++k) {
            v2f a2;
            a2.x = hbuf[wave][lm][4 * k + (hi ? 2 : 0)];
            a2.y = hbuf[wave][lm][4 * k + (hi ? 3 : 1)];
            acc = __builtin_amdgcn_wmma_f32_16x16x4_f32(
                      false, a2, false, bw2[k], (short)0, acc, false, false);
        }

        // Gather proj-mask per output row while all lanes are active.
        float am[8];
#pragma unroll
        for (int v = 0; v < 8; ++v)
            am[v] = wshfl(anym, 16 * t + v + (hi ? 8 : 0));

        // rgb half of output: cols 0..2 live in lanes with (lane&15) < 3.
        if (lm < 3) {
#pragma unroll
            for (int v = 0; v < 8; ++v) {
                int row = v + (hi ? 8 : 0);
                int pp  = base + 16 * t + row;
                if (pp < N)
                    out[(size_t)pp * 6 + lm] = tanhf(acc[v] + b2v) * am[v];
            }
        }
        // WAR: next tile reuses hbuf; keep LDS ops ordered.
        asm volatile("s_wait_dscnt 0" ::: "memory");
    }
}

extern "C" void kernel_launch(void* const* d_in, const int* in_sizes, int n_in,
                              void* d_out, int out_size, void* d_ws, size_t ws_size,
                              hipStream_t stream) {
    const float* images = (const float*)d_in[0];
    const float* extr   = (const float*)d_in[1];
    const float* intr   = (const float*)d_in[2];
    const float* W1     = (const float*)d_in[3];
    const float* b1     = (const float*)d_in[4];
    const float* W2     = (const float*)d_in[5];
    const float* b2     = (const float*)d_in[6];
    const float* pcd    = (const float*)d_in[7];
    const float* scl    = (const float*)d_in[8];
    const int*   dsamp  = (const int*)d_in[9];
    float*       out    = (float*)d_out;

    const int N = in_sizes[7] / 3;              // bg_pcd is (N,3)
    const int blocks = (N + 255) / 256;         // 256 thr = 8 waves = 256 pts

    sky_gaussian_kernel<<<blocks, 256, 0, stream>>>(
        images, extr, intr, W1, b1, W2, b2, pcd, scl, dsamp, out, N);
}